// QRNNLayer_30382598652078
// MI455X (gfx1250) — compile-verified
//
#include <hip/hip_runtime.h>

// ---------------------------------------------------------------------------
// QRNN layer for MI455X (gfx1250, wave32).
//   Y = X @ W^T + b          (M=32768, N=3072, K=1024)  -> full-fp32 WMMA
//   z=tanh, f=sigmoid, o=sigmoid ; scan c_t = z*f + (1-f)*c ; out = o*c
//
// GEMM wave tile: 32M x 64N = 8 accumulators of v_wmma_f32_16x16x4_f32.
// Per k-step (K+=4): 2 A-fragment loads + 4 B-fragment loads feed 8 WMMAs
// (0.75 VMEM issues per WMMA). X (128 MB) and W (12 MB) are L2-resident
// (192 MB L2), and the 8 waves of a block share B addresses via WGP$.
// ---------------------------------------------------------------------------

typedef __attribute__((ext_vector_type(2))) float v2f;
typedef __attribute__((ext_vector_type(8))) float v8f;

#define QB 16
#define QS 2048
#define QD 1024
#define QH 1024
#define QM (QB * QS)   // 32768 rows of the GEMM
#define QN (3 * QH)    // 3072 cols
#define QK QD          // 1024 reduction

__device__ __forceinline__ float sigmoid_f(float x) {
    return 1.0f / (1.0f + expf(-x));
}

// ---------------------------------------------------------------------------
// GEMM + bias + gate activations.
// Block = 256 threads = 8 waves. Block tile = 256 M x 64 N (wave: 32M x 64N).
// ---------------------------------------------------------------------------
__global__ __launch_bounds__(256) void qrnn_gemm_gates(
    const float* __restrict__ X,     // [QM, QK]
    const float* __restrict__ W,     // [QN, QK]
    const float* __restrict__ bias,  // [QN]
    float* __restrict__ zbuf,        // [QM, QH]  tanh(z)
    float* __restrict__ fbuf,        // [QM, QH]  sigmoid(f)
    float* __restrict__ out)         // [QM, QH]  sigmoid(o) (pre-scan)
{
    const int lane = threadIdx.x & 31;
    const int wave = threadIdx.x >> 5;

    const int n0 = blockIdx.x * 64;               // N tile base (64 wide)
    const int m0 = blockIdx.y * 256 + wave * 32;  // M tile base per wave

    // f32 16x16x4 A fragment: lane l -> row (l&15), K = v + 2*(l>>4)
    const int frow  = lane & 15;
    const int khalf = (lane >> 4) * 2;

    const float* Aptr = X + (size_t)(m0 + frow) * QK + khalf;
    const float* Bptr = W + (size_t)(n0 + frow) * QK + khalf;

    // acc[mt*4 + t] : M-subtile mt (0,1), N-subtile t (0..3)
    v8f acc[8];
#pragma unroll
    for (int i = 0; i < 8; ++i) acc[i] = (v8f){0.f,0.f,0.f,0.f,0.f,0.f,0.f,0.f};

#pragma unroll 2
    for (int k = 0; k < QK; k += 4) {
        v2f a0 = *(const v2f*)(Aptr + k);
        v2f a1 = *(const v2f*)(Aptr + (size_t)16 * QK + k);
        v2f b0 = *(const v2f*)(Bptr + k);
        v2f b1 = *(const v2f*)(Bptr + (size_t)16 * QK + k);
        v2f b2 = *(const v2f*)(Bptr + (size_t)32 * QK + k);
        v2f b3 = *(const v2f*)(Bptr + (size_t)48 * QK + k);
        acc[0] = __builtin_amdgcn_wmma_f32_16x16x4_f32(
            false, a0, false, b0, (short)0, acc[0], false, false);
        acc[1] = __builtin_amdgcn_wmma_f32_16x16x4_f32(
            false, a0, false, b1, (short)0, acc[1], false, false);
        acc[2] = __builtin_amdgcn_wmma_f32_16x16x4_f32(
            false, a0, false, b2, (short)0, acc[2], false, false);
        acc[3] = __builtin_amdgcn_wmma_f32_16x16x4_f32(
            false, a0, false, b3, (short)0, acc[3], false, false);
        acc[4] = __builtin_amdgcn_wmma_f32_16x16x4_f32(
            false, a1, false, b0, (short)0, acc[4], false, false);
        acc[5] = __builtin_amdgcn_wmma_f32_16x16x4_f32(
            false, a1, false, b1, (short)0, acc[5], false, false);
        acc[6] = __builtin_amdgcn_wmma_f32_16x16x4_f32(
            false, a1, false, b2, (short)0, acc[6], false, false);
        acc[7] = __builtin_amdgcn_wmma_f32_16x16x4_f32(
            false, a1, false, b3, (short)0, acc[7], false, false);
    }

    // C/D layout: VGPR v, lanes 0-15 -> M = v, lanes 16-31 -> M = v + 8.
    const int nlane = lane & 15;
    const int mhalf = (lane >> 4) * 8;

#pragma unroll
    for (int mt = 0; mt < 2; ++mt) {
#pragma unroll
        for (int t = 0; t < 4; ++t) {
            const int col  = n0 + t * 16 + nlane;    // 0..3071
            const float bv = bias[col];
            const int gate = col >> 10;              // 0=z, 1=f, 2=o
            const int h    = col & (QH - 1);
            const v8f c    = acc[mt * 4 + t];
#pragma unroll
            for (int v = 0; v < 8; ++v) {
                const int m = m0 + mt * 16 + mhalf + v;
                const float val = c[v] + bv;
                const size_t oidx = (size_t)m * QH + h;
                if (gate == 0)      zbuf[oidx] = tanhf(val);
                else if (gate == 1) fbuf[oidx] = sigmoid_f(val);
                else                out[oidx]  = sigmoid_f(val);
            }
        }
    }
}

// ---------------------------------------------------------------------------
// Sequential scan over S per (b,h) channel. One thread per channel; lanes
// across h give fully coalesced 4 KB-strided streams. out[] holds sigmoid(o)
// on entry and is rewritten in place with sigmoid(o)*c. Last c is appended
// at out[QM*QH + b*QH + h]. Loads are s-independent, so unroll-by-4 lets the
// compiler batch 12 outstanding loads ahead of the serial FMA chain.
// ---------------------------------------------------------------------------
__global__ __launch_bounds__(256) void qrnn_scan(
    const float* __restrict__ zbuf,
    const float* __restrict__ fbuf,
    float* __restrict__ out)
{
    const int ch = blockIdx.x * blockDim.x + threadIdx.x;  // 0 .. QB*QH-1
    const int b  = ch >> 10;          // QH == 1024
    const int h  = ch & (QH - 1);
    const size_t base = (size_t)b * QS * QH + h;

    float c = 0.0f;
#pragma unroll 4
    for (int s = 0; s < QS; ++s) {
        const size_t idx = base + (size_t)s * QH;
        const float z  = zbuf[idx];
        const float f  = fbuf[idx];
        const float og = out[idx];
        c = z * f + (1.0f - f) * c;
        out[idx] = og * c;
    }
    out[(size_t)QM * QH + ch] = c;
}

// ---------------------------------------------------------------------------
extern "C" void kernel_launch(void* const* d_in, const int* in_sizes, int n_in,
                              void* d_out, int out_size, void* d_ws, size_t ws_size,
                              hipStream_t stream) {
    const float* X    = (const float*)d_in[0];  // [16, 2048, 1024]
    const float* W    = (const float*)d_in[1];  // [3072, 1024]
    const float* bias = (const float*)d_in[2];  // [3072]
    float* out = (float*)d_out;                 // [16,2048,1024] ++ [16,1,1024]

    float* zbuf = (float*)d_ws;                      // 134 MB
    float* fbuf = zbuf + (size_t)QM * QH;            // 134 MB (268 MB total)

    dim3 grid(QN / 64, QM / 256);                    // 48 x 128 blocks
    qrnn_gemm_gates<<<grid, 256, 0, stream>>>(X, W, bias, zbuf, fbuf, out);

    qrnn_scan<<<(QB * QH) / 256, 256, 0, stream>>>(zbuf, fbuf, out);
}